// GlobalLearnableAttention_88802743812659
// MI455X (gfx1250) — compile-verified
//
#include <hip/hip_runtime.h>
#include <math.h>

// GlobalLearnableAttention for MI455X (gfx1250, wave32).
// Bandwidth-bound (~268 MB total traffic -> ~11.5us at 23.3 TB/s).
// Dot products computed on the matrix pipe with V_WMMA_F32_16X16X4_F32
// (exact f32), 16 batch rows per wave; diagonal of the 16x16 tile gives
// the per-row q.k scores. Streaming blend/normalize phase uses coalesced
// float4 traffic.

typedef __attribute__((ext_vector_type(2))) float v2f;
typedef __attribute__((ext_vector_type(8))) float v8f;

#define DIM 512
#define CHUNKS 128        // DIM / 4 (K=4 per WMMA)
#define ROWS_PER_WAVE 16
#define WAVES_PER_BLOCK 8
#define ROWS_PER_BLOCK (ROWS_PER_WAVE * WAVES_PER_BLOCK)

__global__ __launch_bounds__(WAVES_PER_BLOCK * 32)
void gla_kernel(const float* __restrict__ h0,
                const float* __restrict__ h1,
                const int*   __restrict__ indices,
                const float* __restrict__ Q1,
                const float* __restrict__ K1,
                const float* __restrict__ Q2,
                const float* __restrict__ K2,
                float* __restrict__ out0,
                float* __restrict__ out1)
{
    const int lane = threadIdx.x & 31;
    const int wave = threadIdx.x >> 5;
    const int tileBase = (blockIdx.x * WAVES_PER_BLOCK + wave) * ROWS_PER_WAVE;

    // --- Phase 1: scores via WMMA f32 16x16x4 ---------------------------
    // A layout (16x4 f32): lane<16 holds row (lane&15), K={0,1} in v0/v1;
    // lane>=16 holds same rows, K={2,3}. B (4x16) uses the mirrored layout,
    // so both A and B are loaded with the identical per-lane float2 address.
    const int m    = lane & 15;   // matrix row (= batch row within tile)
    const int half = lane >> 4;   // which K pair of the chunk
    const long idx = (long)indices[tileBase + m];

    const float* q1p = Q1 + idx * DIM + half * 2;
    const float* k1p = K1 + idx * DIM + half * 2;
    const float* q2p = Q2 + idx * DIM + half * 2;
    const float* k2p = K2 + idx * DIM + half * 2;

    v8f acc11 = {}; v8f acc12 = {}; v8f acc21 = {}; v8f acc22 = {};

    #pragma unroll 4
    for (int c = 0; c < CHUNKS; ++c) {
        v2f a1 = *(const v2f*)(q1p + c * 4);
        v2f a2 = *(const v2f*)(q2p + c * 4);
        v2f b1 = *(const v2f*)(k1p + c * 4);
        v2f b2 = *(const v2f*)(k2p + c * 4);
        acc11 = __builtin_amdgcn_wmma_f32_16x16x4_f32(false, a1, false, b1,
                                                      (short)0, acc11, false, false);
        acc12 = __builtin_amdgcn_wmma_f32_16x16x4_f32(false, a1, false, b2,
                                                      (short)0, acc12, false, false);
        acc21 = __builtin_amdgcn_wmma_f32_16x16x4_f32(false, a2, false, b1,
                                                      (short)0, acc21, false, false);
        acc22 = __builtin_amdgcn_wmma_f32_16x16x4_f32(false, a2, false, b2,
                                                      (short)0, acc22, false, false);
    }

    const float scale = 0.04419417382415922f;  // 512^-0.5

    // --- Phase 2+3: per-row softmax, blend, L2-normalize ----------------
    // D layout: VGPR j, lanes 0-15 -> (M=j, N=lane); lanes 16-31 -> (M=8+j,
    // N=lane-16). Diagonal element b sits in VGPR (b&7) at lane b (b<8) or
    // lane b+16 (b>=8).
    #pragma unroll
    for (int b = 0; b < ROWS_PER_WAVE; ++b) {
        const int src = (b < 8) ? b : (b + 16);
        const int reg = b & 7;
        float s11 = __shfl(acc11[reg], src) * scale;
        float s12 = __shfl(acc12[reg], src) * scale;
        float s21 = __shfl(acc21[reg], src) * scale;
        float s22 = __shfl(acc22[reg], src) * scale;

        // 2-way softmax (max-subtracted, like jax.nn.softmax)
        float pm0 = fmaxf(s11, s12);
        float e11 = __expf(s11 - pm0), e12 = __expf(s12 - pm0);
        float r0  = 1.0f / (e11 + e12);
        float p11 = e11 * r0, p12 = e12 * r0;

        float pm1 = fmaxf(s21, s22);
        float e21 = __expf(s21 - pm1), e22 = __expf(s22 - pm1);
        float r1  = 1.0f / (e21 + e22);
        float p21 = e21 * r1, p22 = e22 * r1;

        const long row = tileBase + b;
        const float* h0r = h0 + row * DIM;
        const float* h1r = h1 + row * DIM;

        float4 z0v[4], z1v[4];
        float ss0 = 0.0f, ss1 = 0.0f;
        #pragma unroll
        for (int j = 0; j < 4; ++j) {
            const int cc = (lane + 32 * j) * 4;   // coalesced float4 columns
            float4 a = *(const float4*)(h0r + cc);
            float4 bb = *(const float4*)(h1r + cc);
            float4 z0, z1;
            z0.x = p11 * a.x + p12 * bb.x;  z1.x = p21 * a.x + p22 * bb.x;
            z0.y = p11 * a.y + p12 * bb.y;  z1.y = p21 * a.y + p22 * bb.y;
            z0.z = p11 * a.z + p12 * bb.z;  z1.z = p21 * a.z + p22 * bb.z;
            z0.w = p11 * a.w + p12 * bb.w;  z1.w = p21 * a.w + p22 * bb.w;
            ss0 += z0.x * z0.x + z0.y * z0.y + z0.z * z0.z + z0.w * z0.w;
            ss1 += z1.x * z1.x + z1.y * z1.y + z1.z * z1.z + z1.w * z1.w;
            z0v[j] = z0; z1v[j] = z1;
        }
        // wave32 all-reduce of sum-of-squares
        #pragma unroll
        for (int off = 16; off > 0; off >>= 1) {
            ss0 += __shfl_xor(ss0, off);
            ss1 += __shfl_xor(ss1, off);
        }
        const float inv0 = 1.0f / fmaxf(sqrtf(ss0), 1e-12f);
        const float inv1 = 1.0f / fmaxf(sqrtf(ss1), 1e-12f);

        float* o0 = out0 + row * DIM;
        float* o1 = out1 + row * DIM;
        #pragma unroll
        for (int j = 0; j < 4; ++j) {
            const int cc = (lane + 32 * j) * 4;
            float4 z0 = z0v[j], z1 = z1v[j];
            z0.x *= inv0; z0.y *= inv0; z0.z *= inv0; z0.w *= inv0;
            z1.x *= inv1; z1.y *= inv1; z1.z *= inv1; z1.w *= inv1;
            *(float4*)(o0 + cc) = z0;
            *(float4*)(o1 + cc) = z1;
        }
    }
}

extern "C" void kernel_launch(void* const* d_in, const int* in_sizes, int n_in,
                              void* d_out, int out_size, void* d_ws, size_t ws_size,
                              hipStream_t stream) {
    const float* h0  = (const float*)d_in[0];
    const float* h1  = (const float*)d_in[1];
    const int*   idx = (const int*)d_in[2];
    const float* Q1  = (const float*)d_in[3];
    const float* K1  = (const float*)d_in[4];
    const float* Q2  = (const float*)d_in[5];
    const float* K2  = (const float*)d_in[6];

    const int B = in_sizes[2];          // 16384
    float* out0 = (float*)d_out;        // z0_final, (B, 512)
    float* out1 = out0 + (long)B * DIM; // z1_final, (B, 512)

    dim3 grid(B / ROWS_PER_BLOCK);      // 128 blocks of 8 waves
    dim3 block(WAVES_PER_BLOCK * 32);
    gla_kernel<<<grid, block, 0, stream>>>(h0, h1, idx, Q1, K1, Q2, K2, out0, out1);
}